// GAT_29394756174253
// MI455X (gfx1250) — compile-verified
//
#include <hip/hip_runtime.h>
#include <hip/hip_bf16.h>

// ---------------------------------------------------------------------------
// GAT (2-layer) for MI455X / gfx1250, wave32, V_WMMA_F32_16X16X4_F32 +
// Tensor Data Mover (tensor_load_to_lds) double-buffered weight panels.
// Heavy algebraic fusion: the [2,256,256,512] edge tensor is never stored.
// ---------------------------------------------------------------------------

typedef float v2f __attribute__((ext_vector_type(2)));
typedef float v8f __attribute__((ext_vector_type(8)));
typedef unsigned int u32x4 __attribute__((ext_vector_type(4)));
typedef int i32x4 __attribute__((ext_vector_type(4)));
typedef int i32x8 __attribute__((ext_vector_type(8)));

#define WMMA_F32(a, b, c) \
  __builtin_amdgcn_wmma_f32_16x16x4_f32(false, (a), false, (b), (short)0, (c), false, false)

__device__ __forceinline__ float eluf(float x) {
  return x > 0.f ? x : (__expf(x) - 1.f);
}

__device__ __forceinline__ float blk_max(float v, float* red) {
  const int t = threadIdx.x;
  red[t] = v; __syncthreads();
  for (int s = 128; s > 0; s >>= 1) {
    if (t < s) red[t] = fmaxf(red[t], red[t + s]);
    __syncthreads();
  }
  float r = red[0]; __syncthreads(); return r;
}

__device__ __forceinline__ float blk_sum(float v, float* red) {
  const int t = threadIdx.x;
  red[t] = v; __syncthreads();
  for (int s = 128; s > 0; s >>= 1) {
    if (t < s) red[t] = red[t] + red[t + s];
    __syncthreads();
  }
  float r = red[0]; __syncthreads(); return r;
}

// ---------------------------------------------------------------------------
// TDM: async-load a [16 x 512] f32 tile (contiguous rows, row stride 512
// elements) from global memory into LDS at byte offset lds_off.
// D# layout per CDNA5 ISA ch.8 (group0: 4 dwords, group1: 8 dwords).
// ---------------------------------------------------------------------------
__device__ __forceinline__ void tdm_load_panel16x512(const float* gsrc,
                                                     unsigned int lds_off) {
  const unsigned long long ga = (unsigned long long)(uintptr_t)gsrc;
  u32x4 g0;
  g0[0] = 1u;                                   // count=1 (valid), user mode
  g0[1] = lds_off;                              // lds_addr (bytes)
  g0[2] = (unsigned int)(ga & 0xffffffffu);     // global_addr[31:0]
  g0[3] = (unsigned int)((ga >> 32) & 0x1ffffffu) | (2u << 30); // [56:32]|type=2
  i32x8 g1;
  g1[0] = 0x00020000;          // workgroup_mask=0, data_size=2 (4 bytes)
  g1[1] = (int)(512u << 16);   // tensor_dim0[15:0]=512 in bits[31:16]
  g1[2] = (int)(512u << 16);   // tensor_dim0 hi=0 | tensor_dim1[15:0]=512
  g1[3] = (int)(512u << 16);   // tensor_dim1 hi=0 | tile_dim0=512
  g1[4] = 16;                  // tile_dim1=16 rows, tile_dim2=0
  g1[5] = 512;                 // tensor_dim0_stride[31:0]=512
  g1[6] = 0;                   // stride0 hi | tensor_dim1_stride lo
  g1[7] = 0;
  const i32x4 z = {0, 0, 0, 0};
#if defined(__clang_major__) && (__clang_major__ >= 23)
  const i32x8 z8 = {0, 0, 0, 0, 0, 0, 0, 0};
  __builtin_amdgcn_tensor_load_to_lds(g0, g1, z, z, z8, 0);
#else
  __builtin_amdgcn_tensor_load_to_lds(g0, g1, z, z, 0);
#endif
}

// ---------------------------------------------------------------------------
// Generic tiled WMMA f32 GEMM: C[M,N] = A[M,K] @ B[K,N] (+ bias[N]).
// grid = (M/16, ceil(N/128)), block = 256 (8 waves, one 16x16 tile each).
// ---------------------------------------------------------------------------
__global__ void __launch_bounds__(256) k_gemm(
    const float* __restrict__ A, int lda,
    const float* __restrict__ B, int ldb,
    const float* __restrict__ bias,
    float* __restrict__ C, int ldc,
    int M, int N, int K)
{
  __shared__ float As[16][36];
  __shared__ float Bs[32][132];
  const int tid  = threadIdx.x;
  const int wave = tid >> 5, lane = tid & 31;
  const int am   = lane & 15;
  const int kh   = (lane >> 4) << 1;   // 0 or 2 (K sub-offset)
  const int rh   = (lane >> 4) << 3;   // 0 or 8 (M sub-offset for C frag)
  const int m0   = blockIdx.x * 16;
  const int nblk = blockIdx.y * 128;
  const int ncol = nblk + wave * 16 + am;

  v8f acc;
  {
    float bv = (bias != nullptr && ncol < N) ? bias[ncol] : 0.f;
#pragma unroll
    for (int r = 0; r < 8; ++r) acc[r] = bv;
  }

  for (int k0 = 0; k0 < K; k0 += 32) {
    for (int idx = tid; idx < 16 * 32; idx += 256) {
      int r = idx >> 5, kk = idx & 31;
      As[r][kk] = (k0 + kk < K && m0 + r < M)
                      ? A[(size_t)(m0 + r) * lda + k0 + kk] : 0.f;
    }
    for (int idx = tid; idx < 32 * 128; idx += 256) {
      int r = idx >> 7, c = idx & 127;
      Bs[r][c] = (k0 + r < K && nblk + c < N)
                     ? B[(size_t)(k0 + r) * ldb + nblk + c] : 0.f;
    }
    __syncthreads();
#pragma unroll
    for (int kk = 0; kk < 32; kk += 4) {
      v2f a, bb;
      a.x  = As[am][kk + kh];
      a.y  = As[am][kk + kh + 1];
      bb.x = Bs[kk + kh][wave * 16 + am];
      bb.y = Bs[kk + kh + 1][wave * 16 + am];
      acc  = WMMA_F32(a, bb, acc);
    }
    __syncthreads();
  }

  if (ncol < N) {
#pragma unroll
    for (int r = 0; r < 8; ++r) {
      int row = m0 + r + rh;
      if (row < M) C[(size_t)row * ldc + ncol] = acc[r];
    }
  }
}

// ---------------------------------------------------------------------------
// Node update: out[b,x,c] = elu(nf[b,x,c] + sum_i an[b,i,x]*nf[b,i,c]
//                               + eacc[b,x,c]) (+ addvec[b,c])
// A is an^T (strided loads). M=256, N=512, K=256 per batch.
// grid = (16, 4, B), block 256.
// ---------------------------------------------------------------------------
__global__ void __launch_bounds__(256) k_nodeupd(
    const float* __restrict__ an, const float* __restrict__ nf,
    const float* __restrict__ eacc, const float* __restrict__ addvec,
    float* __restrict__ outN)
{
  const int b    = blockIdx.z;
  const int m0   = blockIdx.x * 16;
  const int nblk = blockIdx.y * 128;
  const int tid  = threadIdx.x;
  const int wave = tid >> 5, lane = tid & 31;
  const int am   = lane & 15;
  const int kh   = (lane >> 4) << 1;
  const int rh   = (lane >> 4) << 3;

  __shared__ float As[16][36];
  __shared__ float Bs[32][132];

  const float* anb = an + (size_t)b * 65536;
  const float* nfb = nf + (size_t)b * 131072;

  v8f acc;
#pragma unroll
  for (int r = 0; r < 8; ++r) acc[r] = 0.f;

  for (int k0 = 0; k0 < 256; k0 += 32) {
    for (int idx = tid; idx < 16 * 32; idx += 256) {
      int r = idx >> 5, kk = idx & 31;
      As[r][kk] = anb[(size_t)(k0 + kk) * 256 + m0 + r];   // transposed A
    }
    for (int idx = tid; idx < 32 * 128; idx += 256) {
      int r = idx >> 7, c = idx & 127;
      Bs[r][c] = nfb[(size_t)(k0 + r) * 512 + nblk + c];
    }
    __syncthreads();
#pragma unroll
    for (int kk = 0; kk < 32; kk += 4) {
      v2f a, bb;
      a.x  = As[am][kk + kh];
      a.y  = As[am][kk + kh + 1];
      bb.x = Bs[kk + kh][wave * 16 + am];
      bb.y = Bs[kk + kh + 1][wave * 16 + am];
      acc  = WMMA_F32(a, bb, acc);
    }
    __syncthreads();
  }

  const int n = nblk + wave * 16 + am;
#pragma unroll
  for (int r = 0; r < 8; ++r) {
    const int row = m0 + r + rh;
    float v = nfb[(size_t)row * 512 + n] + acc[r] +
              eacc[(size_t)b * 131072 + (size_t)row * 512 + n];
    v = eluf(v);
    if (addvec != nullptr) v += addvec[b * 512 + n];
    outN[(size_t)b * 131072 + (size_t)row * 512 + n] = v;
  }
}

// ---------------------------------------------------------------------------
// Small kernels (VALU): conditioning, mask, folded-weight precompute, q-proj.
// ---------------------------------------------------------------------------
__global__ void k_cond(const float* __restrict__ te,
                       const float* __restrict__ Wnc, const float* __restrict__ bnc,
                       const float* __restrict__ Wec, const float* __restrict__ bec,
                       float* __restrict__ ten, float* __restrict__ tee)
{
  const int b = blockIdx.x;
  for (int c = threadIdx.x; c < 512; c += blockDim.x) {
    float s1 = bnc[c], s2 = bec[c];
    for (int k = 0; k < 129; ++k) {
      float v = te[b * 129 + k];
      s1 += v * Wnc[k * 512 + c];
      s2 += v * Wec[k * 512 + c];
    }
    ten[b * 512 + c] = fmaxf(s1, 0.f);
    tee[b * 512 + c] = fmaxf(s2, 0.f);
  }
}

__global__ void k_mask(const float* __restrict__ nodes, float* __restrict__ maskf)
{
  const int bi = blockIdx.x * 256 + threadIdx.x;   // b*256+i
  maskf[bi] = (nodes[(size_t)bi * 16 + 15] == 0.f) ? 1.f : 0.f;
}

// M0[e,h] = sum_c We0[e,c]*Wa0[c,h]; m0b[h] = sum_c be0[c]*Wa0[c,h];
// m1b[h] = sum_c be1[c]*Wa1[c,h].   (one block, 128 threads)
__global__ void k_fold0(const float* __restrict__ We0, const float* __restrict__ be0,
                        const float* __restrict__ Wa0,
                        const float* __restrict__ be1, const float* __restrict__ Wa1,
                        float* __restrict__ M0, float* __restrict__ m0b,
                        float* __restrict__ m1b)
{
  const int t = threadIdx.x;
  if (t < 64) {
    int e = t >> 3, h = t & 7;
    float s = 0.f;
    for (int c = 0; c < 512; ++c) s += We0[e * 512 + c] * Wa0[c * 8 + h];
    M0[t] = s;
  } else if (t < 72) {
    int h = t - 64; float s = 0.f;
    for (int c = 0; c < 512; ++c) s += be0[c] * Wa0[c * 8 + h];
    m0b[h] = s;
  } else if (t < 80) {
    int h = t - 72; float s = 0.f;
    for (int c = 0; c < 512; ++c) s += be1[c] * Wa1[c * 8 + h];
    m1b[h] = s;
  }
}

// P1[c,h] = sum_c' We1[c,c'] * Wa1[c',h]  (4096 outputs; grid 16 x 256)
__global__ void k_fold1(const float* __restrict__ We1, const float* __restrict__ Wa1,
                        float* __restrict__ P1)
{
  const int gid = blockIdx.x * 256 + threadIdx.x;
  const int c = gid >> 3, h = gid & 7;
  float s = 0.f;
  for (int cp = 0; cp < 512; ++cp) s += We1[(size_t)c * 512 + cp] * Wa1[cp * 8 + h];
  P1[gid] = s;
}

// q[row,h] = sum_c X[row,c]*Wa[c,h]   (rows*8 threads total)
__global__ void k_qproj(const float* __restrict__ X, const float* __restrict__ Wa,
                        float* __restrict__ q, int rows)
{
  const int gid = blockIdx.x * 256 + threadIdx.x;
  if (gid >= rows * 8) return;
  const int r = gid >> 3, h = gid & 7;
  float s = 0.f;
  for (int c = 0; c < 512; ++c) s += X[(size_t)r * 512 + c] * Wa[c * 8 + h];
  q[gid] = s;
}

// u1[bi,c] = sum_h q1[bi,h]*P1[c,h]   (grid 1024 x 256)
__global__ void k_u1(const float* __restrict__ q1, const float* __restrict__ P1,
                     float* __restrict__ u1)
{
  const int gid = blockIdx.x * 256 + threadIdx.x;
  const int row = gid >> 9, c = gid & 511;
  float s = 0.f;
#pragma unroll
  for (int h = 0; h < 8; ++h) s += q1[row * 8 + h] * P1[c * 8 + h];
  u1[gid] = s;
}

// ---------------------------------------------------------------------------
// Layer-0 scores + softmax: one block per (b,i) row; thread j owns column j.
// sn = q0_i . q0_j / 8 ; se = (edge_raw . v0 + c0)/8 ; softmax with row mask.
// ---------------------------------------------------------------------------
__global__ void __launch_bounds__(256) k_score_soft0(
    const float* __restrict__ q0, const float* __restrict__ edges,
    const float* __restrict__ M0, const float* __restrict__ m0b,
    const float* __restrict__ maskf,
    float* __restrict__ an0, float* __restrict__ ae0)
{
  const int bi = blockIdx.x;
  const int b  = bi >> 8;
  const int j  = threadIdx.x;
  __shared__ float qi[8], v0[8], red[256];
  __shared__ float c0s;
  if (j < 8) qi[j] = q0[bi * 8 + j];
  __syncthreads();
  if (j < 8) {
    float s = 0.f;
#pragma unroll
    for (int h = 0; h < 8; ++h) s += qi[h] * M0[j * 8 + h];
    v0[j] = s;
  }
  if (j == 8) {
    float s = 0.f;
#pragma unroll
    for (int h = 0; h < 8; ++h) s += qi[h] * m0b[h];
    c0s = s;
  }
  __syncthreads();

  float sn = 0.f;
#pragma unroll
  for (int h = 0; h < 8; ++h) sn += qi[h] * q0[(b * 256 + j) * 8 + h];
  const float* ed = edges + ((size_t)bi * 256 + j) * 8;
  float se = c0s;
#pragma unroll
  for (int e = 0; e < 8; ++e) se += ed[e] * v0[e];

  const float m = maskf[bi];
  sn *= 0.125f * m;
  se *= 0.125f * m;

  float mx = blk_max(sn, red);
  float p  = __expf(sn - mx);
  float sm = blk_sum(p, red);
  an0[(size_t)bi * 256 + j] = p / sm;

  mx = blk_max(se, red);
  p  = __expf(se - mx);
  sm = blk_sum(p, red);
  ae0[(size_t)bi * 256 + j] = p / sm;
}

// ---------------------------------------------------------------------------
// eacc0[b,i,:] = (sum_j ae0[b,i,j]*edge_raw[b,i,j,:]) @ We0 + be0
// ---------------------------------------------------------------------------
__global__ void __launch_bounds__(256) k_r0_eacc0(
    const float* __restrict__ edges, const float* __restrict__ ae0,
    const float* __restrict__ We0, const float* __restrict__ be0,
    float* __restrict__ eacc0)
{
  const int bi = blockIdx.x;
  const int j  = threadIdx.x;
  __shared__ float part[256][9];
  __shared__ float r0[8];
  const float a = ae0[(size_t)bi * 256 + j];
  const float* ed = edges + ((size_t)bi * 256 + j) * 8;
#pragma unroll
  for (int e = 0; e < 8; ++e) part[j][e] = a * ed[e];
  __syncthreads();
  if (j < 8) {
    float s = 0.f;
    for (int t = 0; t < 256; ++t) s += part[t][j];
    r0[j] = s;
  }
  __syncthreads();
  for (int c = j; c < 512; c += 256) {
    float s = be0[c];
#pragma unroll
    for (int e = 0; e < 8; ++e) s += r0[e] * We0[e * 512 + c];
    eacc0[(size_t)bi * 512 + c] = s;
  }
}

// ---------------------------------------------------------------------------
// Layer-1 scores + softmax. se1[b,i,j] = (e0_row[j,:] . u1[b,i,:] + c1)/8 with
// e0_row recomputed on the fly from raw 8-dim edges (layer-0 output edges).
// ---------------------------------------------------------------------------
__global__ void __launch_bounds__(256) k_score_soft1(
    const float* __restrict__ q1, const float* __restrict__ edges,
    const float* __restrict__ We0, const float* __restrict__ be0,
    const float* __restrict__ nf0, const float* __restrict__ an0,
    const float* __restrict__ ae0, const float* __restrict__ tee,
    const float* __restrict__ u1, const float* __restrict__ m1b,
    const float* __restrict__ maskf,
    float* __restrict__ an1, float* __restrict__ ae1)
{
  const int bi = blockIdx.x;
  const int b  = bi >> 8;
  const int j  = threadIdx.x;
  __shared__ float We0s[8][513];
  __shared__ float be0s[512], nf0s[512], tees[512], u1s[512];
  __shared__ float q1i[8], red[256];
  __shared__ float c1s;

  for (int idx = j; idx < 8 * 512; idx += 256) We0s[idx >> 9][idx & 511] = We0[idx];
  for (int idx = j; idx < 512; idx += 256) {
    be0s[idx] = be0[idx];
    nf0s[idx] = nf0[(size_t)bi * 512 + idx];
    tees[idx] = tee[b * 512 + idx];
    u1s[idx]  = u1[(size_t)bi * 512 + idx];
  }
  if (j < 8) q1i[j] = q1[bi * 8 + j];
  __syncthreads();
  if (j == 0) {
    float s = 0.f;
#pragma unroll
    for (int h = 0; h < 8; ++h) s += q1i[h] * m1b[h];
    c1s = s;
  }
  __syncthreads();

  const float ae0j = ae0[(size_t)bi * 256 + j];
  const float an0j = an0[(size_t)bi * 256 + j];
  float e8[8];
  {
    const float* ed = edges + ((size_t)bi * 256 + j) * 8;
#pragma unroll
    for (int e = 0; e < 8; ++e) e8[e] = ed[e];
  }

  float se = c1s;
  for (int c = 0; c < 512; ++c) {
    float ef0 = be0s[c];
#pragma unroll
    for (int e = 0; e < 8; ++e) ef0 += e8[e] * We0s[e][c];
    float t = ef0 * (1.f + ae0j) + an0j * nf0s[c];
    t = eluf(t) + tees[c];
    se += t * u1s[c];
  }

  float sn = 0.f;
#pragma unroll
  for (int h = 0; h < 8; ++h) sn += q1i[h] * q1[(b * 256 + j) * 8 + h];

  const float m = maskf[bi];
  sn *= 0.125f * m;
  se *= 0.125f * m;

  float mx = blk_max(sn, red);
  float p  = __expf(sn - mx);
  float sm = blk_sum(p, red);
  an1[(size_t)bi * 256 + j] = p / sm;

  mx = blk_max(se, red);
  p  = __expf(se - mx);
  sm = blk_sum(p, red);
  ae1[(size_t)bi * 256 + j] = p / sm;
}

// ---------------------------------------------------------------------------
// THE fused edge kernel (layer 1 + final projection). One block per (b,i):
//   per 16-row j-tile: recompute e0 tile in LDS -> WMMA GEMM ef1 = e0 @ We1
//   with TDM double-buffered We1 k-panels (tensor_load_to_lds overlapped with
//   WMMA, TENSORcnt-synced) -> elementwise update + ELU (accumulate r1) ->
//   out_edges tile = upd @ Wfe + bfe.  The 256 MB ef tensor never hits HBM.
// ---------------------------------------------------------------------------
__global__ void __launch_bounds__(256) k_edge1(
    const float* __restrict__ edges,
    const float* __restrict__ We0, const float* __restrict__ be0,
    const float* __restrict__ We1, const float* __restrict__ be1,
    const float* __restrict__ Wfe, const float* __restrict__ bfe,
    const float* __restrict__ nf0, const float* __restrict__ nf1,
    const float* __restrict__ tee,
    const float* __restrict__ an0, const float* __restrict__ ae0,
    const float* __restrict__ an1, const float* __restrict__ ae1,
    float* __restrict__ out_edges, float* __restrict__ r1)
{
  const int bi   = blockIdx.x;
  const int b    = bi >> 8;
  const int tid  = threadIdx.x;
  const int wave = tid >> 5, lane = tid & 31;
  const int am   = lane & 15;
  const int kh   = (lane >> 4) << 1;
  const int rh   = (lane >> 4) << 3;

  __shared__ float We0s[8][513];
  __shared__ float be0s[512];
  __shared__ float nf0s[512], nf1s[512], tees[512], be1s[512];
  __shared__ float ae0r[256], an0r[256], ae1r[256], an1r[256];
  __shared__ float e0t[16][516];
  __shared__ float ef1t[16][516];
  __shared__ float wpan[2][16][512];   // TDM double buffer (contiguous rows)
  __shared__ float edg[16][8];
  __shared__ float bfes[8];

  for (int idx = tid; idx < 8 * 512; idx += 256) We0s[idx >> 9][idx & 511] = We0[idx];
  for (int idx = tid; idx < 512; idx += 256) {
    be0s[idx] = be0[idx];
    be1s[idx] = be1[idx];
    nf0s[idx] = nf0[(size_t)bi * 512 + idx];
    nf1s[idx] = nf1[(size_t)bi * 512 + idx];
    tees[idx] = tee[b * 512 + idx];
  }
  ae0r[tid] = ae0[(size_t)bi * 256 + tid];
  an0r[tid] = an0[(size_t)bi * 256 + tid];
  ae1r[tid] = ae1[(size_t)bi * 256 + tid];
  an1r[tid] = an1[(size_t)bi * 256 + tid];
  if (tid < 8) bfes[tid] = bfe[tid];

  // LDS byte offsets of the two TDM panel buffers
  const unsigned int wpan_off0 =
      (unsigned int)(uintptr_t)(&wpan[0][0][0]);
  const unsigned int wpan_off1 =
      (unsigned int)(uintptr_t)(&wpan[1][0][0]);

  float racc0 = 0.f, racc1 = 0.f;            // r1 columns 2*tid, 2*tid+1
  const int cA = 2 * tid, cB = 2 * tid + 1;
  __syncthreads();

  for (int jt = 0; jt < 16; ++jt) {
    const int j0 = jt * 16;

    if (tid < 128) {
      int jj = tid >> 3, e = tid & 7;
      edg[jj][e] = edges[((size_t)bi * 256 + j0 + jj) * 8 + e];
    }
    __syncthreads();

    // (a) recompute layer-0 output edge tile e0t[16][512]
    for (int idx = tid; idx < 16 * 512; idx += 256) {
      int jj = idx >> 9, c = idx & 511;
      float s = be0s[c];
#pragma unroll
      for (int e = 0; e < 8; ++e) s += edg[jj][e] * We0s[e][c];
      float t = s * (1.f + ae0r[j0 + jj]) + an0r[j0 + jj] * nf0s[c];
      e0t[jj][c] = eluf(t) + tees[c];
    }

    // (b) ef1 tile = e0t @ We1 + be1 (WMMA f32, TDM double-buffered panels)
    v8f acc[4];
#pragma unroll
    for (int st = 0; st < 4; ++st) {
      float bv = be1s[wave * 64 + st * 16 + am];
#pragma unroll
      for (int r = 0; r < 8; ++r) acc[st][r] = bv;
    }
    // prologue: DMA panel 0 into buffer 0 (wave 0 issues, all waves sync)
    if (wave == 0) {
      tdm_load_panel16x512(We1, wpan_off0);
      __builtin_amdgcn_s_wait_tensorcnt(0);
    }
    __syncthreads();   // e0t ready + panel 0 ready

    int buf = 0;
    for (int k0 = 0; k0 < 512; k0 += 16) {
      const bool more = (k0 + 16) < 512;
      if (more && wave == 0)   // overlap: DMA next panel while computing
        tdm_load_panel16x512(We1 + (size_t)(k0 + 16) * 512,
                             buf ? wpan_off0 : wpan_off1);
#pragma unroll
      for (int kk = 0; kk < 16; kk += 4) {
        v2f a;
        a.x = e0t[am][k0 + kk + kh];
        a.y = e0t[am][k0 + kk + kh + 1];
#pragma unroll
        for (int st = 0; st < 4; ++st) {
          v2f bb;
          const int n = wave * 64 + st * 16 + am;
          bb.x = wpan[buf][kk + kh][n];
          bb.y = wpan[buf][kk + kh + 1][n];
          acc[st] = WMMA_F32(a, bb, acc[st]);
        }
      }
      __syncthreads();                 // all reads of wpan[buf] complete
      if (more && wave == 0)
        __builtin_amdgcn_s_wait_tensorcnt(0);
      __syncthreads();                 // next panel visible to all waves
      buf ^= 1;
    }
#pragma unroll
    for (int st = 0; st < 4; ++st) {
      const int n = wave * 64 + st * 16 + am;
#pragma unroll
      for (int r = 0; r < 8; ++r) ef1t[r + rh][n] = acc[st][r];
    }
    __syncthreads();

    // (c) elementwise update + ELU; accumulate r1 = sum_j ae1[j]*e0[j,:]
#pragma unroll 4
    for (int jj = 0; jj < 16; ++jj) {
      const float aej = ae1r[j0 + jj], anj = an1r[j0 + jj];
      float vA = ef1t[jj][cA] * (1.f + aej) + anj * nf1s[cA];
      float vB = ef1t[jj][cB] * (1.f + aej) + anj * nf1s[cB];
      racc0 += aej * e0t[jj][cA];
      racc1 += aej * e0t[jj][cB];
      ef1t[jj][cA] = eluf(vA);
      ef1t[jj][cB] = eluf(vB);
    }
    __syncthreads();

    // (d) out_edges tile [16,8] = upd @ Wfe + bfe
    if (tid < 128) {
      const int jj = tid >> 3, o = tid & 7;
      float s = bfes[o];
      for (int c = 0; c < 512; ++c) s += ef1t[jj][c] * Wfe[c * 8 + o];
      out_edges[((size_t)bi * 256 + j0 + jj) * 8 + o] = s;
    }
    __syncthreads();
  }

  r1[(size_t)bi * 512 + cA] = racc0;
  r1[(size_t)bi * 512 + cB] = racc1;
}

// ---------------------------------------------------------------------------
// Host-side launch orchestration.
// ---------------------------------------------------------------------------
extern "C" void kernel_launch(void* const* d_in, const int* in_sizes, int n_in,
                              void* d_out, int out_size, void* d_ws, size_t ws_size,
                              hipStream_t stream)
{
  (void)in_sizes; (void)n_in; (void)out_size; (void)ws_size;

  // setup_inputs() insertion order, params recursed in insertion order.
  const float* nodes = (const float*)d_in[0];   // [2,256,16]
  const float* edges = (const float*)d_in[1];   // [2,256,256,8]
  const float* te    = (const float*)d_in[2];   // [2,129]
  const float* Wn0   = (const float*)d_in[3];   // [16,512]
  const float* bn0   = (const float*)d_in[4];
  const float* We0   = (const float*)d_in[5];   // [8,512]
  const float* be0   = (const float*)d_in[6];
  const float* Wa0   = (const float*)d_in[7];   // [512,8]
  const float* Wn1   = (const float*)d_in[8];   // [512,512]
  const float* bn1   = (const float*)d_in[9];
  const float* We1   = (const float*)d_in[10];  // [512,512]
  const float* be1   = (const float*)d_in[11];
  const float* Wa1   = (const float*)d_in[12];  // [512,8]
  const float* Wnc   = (const float*)d_in[13];  // [129,512]
  const float* bnc   = (const float*)d_in[14];
  const float* Wec   = (const float*)d_in[15];  // [129,512]
  const float* bec   = (const float*)d_in[16];
  const float* Wfn   = (const float*)d_in[17];  // [512,16]
  const float* bfn   = (const float*)d_in[18];
  const float* Wfe   = (const float*)d_in[19];  // [512,8]
  const float* bfe   = (const float*)d_in[20];

  float* Wk = (float*)d_ws;
  size_t off = 0;
  auto wsa = [&](size_t n) { float* p = Wk + off; off += n; return p; };

  float* ten   = wsa(1024);       // [2,512]
  float* tee   = wsa(1024);       // [2,512]
  float* maskf = wsa(512);        // [2,256]
  float* M0w   = wsa(64);         // [8,8]
  float* m0bw  = wsa(8);
  float* m1bw  = wsa(8);
  float* P1w   = wsa(4096);       // [512,8]
  float* nf0   = wsa(262144);     // [2,256,512]
  float* q0    = wsa(4096);       // [2,256,8]
  float* an0   = wsa(131072);     // [2,256,256]
  float* ae0   = wsa(131072);
  float* eacc0 = wsa(262144);
  float* n0out = wsa(262144);
  float* nf1   = wsa(262144);
  float* q1    = wsa(4096);
  float* u1    = wsa(262144);
  float* an1   = wsa(131072);
  float* ae1   = wsa(131072);
  float* r1w   = wsa(262144);
  float* eacc1 = wsa(262144);
  float* n1out = wsa(262144);

  float* outNodes = (float*)d_out;              // [2,256,16]
  float* outEdges = (float*)d_out + 8192;       // [2,256,256,8]

  // preprocessing
  k_cond <<<2, 256, 0, stream>>>(te, Wnc, bnc, Wec, bec, ten, tee);
  k_mask <<<2, 256, 0, stream>>>(nodes, maskf);
  k_fold0<<<1, 128, 0, stream>>>(We0, be0, Wa0, be1, Wa1, M0w, m0bw, m1bw);
  k_fold1<<<16, 256, 0, stream>>>(We1, Wa1, P1w);

  // ---- layer 0 ----
  k_gemm<<<dim3(32, 4), 256, 0, stream>>>(nodes, 16, Wn0, 512, bn0,
                                          nf0, 512, 512, 512, 16);
  k_qproj<<<16, 256, 0, stream>>>(nf0, Wa0, q0, 512);
  k_score_soft0<<<512, 256, 0, stream>>>(q0, edges, M0w, m0bw, maskf, an0, ae0);
  k_r0_eacc0<<<512, 256, 0, stream>>>(edges, ae0, We0, be0, eacc0);
  k_nodeupd<<<dim3(16, 4, 2), 256, 0, stream>>>(an0, nf0, eacc0, ten, n0out);

  // ---- layer 1 nodes / scores ----
  k_gemm<<<dim3(32, 4), 256, 0, stream>>>(n0out, 512, Wn1, 512, bn1,
                                          nf1, 512, 512, 512, 512);
  k_qproj<<<16, 256, 0, stream>>>(nf1, Wa1, q1, 512);
  k_u1<<<1024, 256, 0, stream>>>(q1, P1w, u1);
  k_score_soft1<<<512, 256, 0, stream>>>(q1, edges, We0, be0, nf0, an0, ae0,
                                         tee, u1, m1bw, maskf, an1, ae1);

  // ---- fused layer-1 edges + final edge projection ----
  k_edge1<<<512, 256, 0, stream>>>(edges, We0, be0, We1, be1, Wfe, bfe,
                                   nf0, nf1, tee, an0, ae0, an1, ae1,
                                   outEdges, r1w);

  // ---- layer-1 node update + final node projection ----
  k_gemm<<<dim3(32, 4), 256, 0, stream>>>(r1w, 512, We1, 512, be1,
                                          eacc1, 512, 512, 512, 512);
  k_nodeupd<<<dim3(16, 4, 2), 256, 0, stream>>>(an1, nf1, eacc1, nullptr, n1out);
  k_gemm<<<dim3(32, 1), 256, 0, stream>>>(n1out, 512, Wfn, 16, bfn,
                                          outNodes, 16, 512, 16, 512);
}